// ChannelWiseQuantumCircuit_70978629533755
// MI455X (gfx1250) — compile-verified
//
#include <hip/hip_runtime.h>
#include <hip/hip_bf16.h>

// ---------------------------------------------------------------------------
// 12-qubit channel-wise quantum circuit simulator for MI455X (gfx1250).
//
// Strategy (compute-bound, state fits LDS):
//  * 1 workgroup (256 thr = 8 wave32) per batch element; state (4096 c64)
//    lives in LDS as split re[4096]/im[4096] float arrays (32 KB).
//  * Per patch position: fuse RZ*RY*RX (weights) with RY(data) into one 2x2
//    complex gate per qubit; fuse 3 qubits -> 8x8 complex = 16x16 REAL matrix
//    U^ = [[Re,-Im],[Im,Re]]; apply as real GEMM 16x16 x 16x512 using
//    V_WMMA_F32_16X16X4_F32 (M=16, K=16 via 4 slices, N=512 via 32 tiles).
//  * The CX ring (CX(q,q+1) chain + CX(11,0)) is a pure GF(2) basis
//    permutation: new bit j = suffix-XOR of old bits -> single scatter pass.
//  * <Z_i> from per-thread partial sums + LDS float atomics.
//
// Bit convention: amp-index bit j (LSB=0) == reference qubit (11-j)
// (row-major flattening of the (2,)*12 jnp array).
// ---------------------------------------------------------------------------

typedef __attribute__((ext_vector_type(2))) float v2f;
typedef __attribute__((ext_vector_type(8))) float v8f;

#define NQ        12
#define NPOS      9
#define NAMP      4096       // 2^12
#define THREADS   256
#define AMP_SCALE 0.015625f  // 2^-6 uniform superposition amplitude

__global__ __launch_bounds__(THREADS)
void qcircuit_sim_kernel(const float* __restrict__ weights,   // (9,1,12,3) flat
                         const float* __restrict__ patches,   // (B,12,9) flat
                         float* __restrict__ out)             // (B,12)
{
    // LDS: S[0..4095]=Re(state), S[4096..8191]=Im(state)
    __shared__ float S[2 * NAMP];
    __shared__ float Uhat[16][17];   // padded to dodge bank conflicts
    __shared__ float Gm[NQ][8];      // fused 2x2 complex gate per (my) bit
    __shared__ float zacc[NQ];

    const int tid  = threadIdx.x;
    const int b    = blockIdx.x;
    const int lane = tid & 31;
    const int wave = tid >> 5;

    // ---- init |+>^12 : all amplitudes 2^-6 + 0i ----
    #pragma unroll
    for (int k = 0; k < 16; ++k) {
        int i = tid + THREADS * k;
        S[i]        = AMP_SCALE;
        S[NAMP + i] = 0.0f;
    }
    __syncthreads();

    for (int pos = 0; pos < NPOS; ++pos) {
        // ---- build 12 fused 1-qubit gates: G = RZ(w2)*RY(w1)*RX(w0)*RY(th) ----
        if (tid < NQ) {
            const int j = tid;        // my bit
            const int q = 11 - j;     // reference qubit / channel
            const int wbase = (pos * NQ + q) * 3;
            float sx, cx, sy, cy, sz, cz, st, ct;
            __sincosf(0.5f * weights[wbase + 0], &sx, &cx);   // RX
            __sincosf(0.5f * weights[wbase + 1], &sy, &cy);   // RY
            __sincosf(0.5f * weights[wbase + 2], &sz, &cz);   // RZ
            __sincosf(0.5f * patches[(b * NQ + q) * NPOS + pos], &st, &ct); // data RY
            // M1 = RY*RX
            float m00r =  cy * cx, m00i =  sy * sx;
            float m01r = -sy * cx, m01i = -cy * sx;
            float m10r =  sy * cx, m10i = -cy * sx;
            float m11r =  cy * cx, m11i = -sy * sx;
            // W = RZ*M1 : row0 *= (cz - i sz), row1 *= (cz + i sz)
            float W00r = cz*m00r + sz*m00i, W00i = cz*m00i - sz*m00r;
            float W01r = cz*m01r + sz*m01i, W01i = cz*m01i - sz*m01r;
            float W10r = cz*m10r - sz*m10i, W10i = cz*m10i + sz*m10r;
            float W11r = cz*m11r - sz*m11i, W11i = cz*m11i + sz*m11r;
            // G = W * RY(th),  RY(th) = [[ct,-st],[st,ct]] (real)
            Gm[j][0] =  W00r*ct + W01r*st;  Gm[j][1] =  W00i*ct + W01i*st;   // G00
            Gm[j][2] = -W00r*st + W01r*ct;  Gm[j][3] = -W00i*st + W01i*ct;   // G01
            Gm[j][4] =  W10r*ct + W11r*st;  Gm[j][5] =  W10i*ct + W11i*st;   // G10
            Gm[j][6] = -W10r*st + W11r*ct;  Gm[j][7] = -W10i*st + W11i*ct;   // G11
        }
        __syncthreads();

        // ---- 4 groups of 3 qubits; each = 16x16 real matmul over the state ----
        for (int g = 0; g < 4; ++g) {
            // Build Uhat (16x16 real form of 8x8 complex tensor product):
            // one entry per thread.
            {
                const int m = tid >> 4, k = tid & 15;
                const int mr = m & 7,  kr = k & 7;
                const float* G0 = Gm[3*g + 0];
                const float* G1 = Gm[3*g + 1];
                const float* G2 = Gm[3*g + 2];
                const int e2 = (((mr>>2)&1)*2 + ((kr>>2)&1)) * 2;
                const int e1 = (((mr>>1)&1)*2 + ((kr>>1)&1)) * 2;
                const int e0 = (((mr   )&1)*2 + ((kr   )&1)) * 2;
                float ar = G2[e2], ai = G2[e2+1];
                float br = G1[e1], bi = G1[e1+1];
                float tr = ar*br - ai*bi, ti = ar*bi + ai*br;
                br = G0[e0]; bi = G0[e0+1];
                float ur = tr*br - ti*bi;     // Re U[mr][kr]
                float ui = tr*bi + ti*br;     // Im U[mr][kr]
                // [[Re,-Im],[Im,Re]]
                Uhat[m][k] = (m < 8) ? ((k < 8) ? ur : -ui)
                                     : ((k < 8) ? ui :  ur);
            }
            __syncthreads();

            // A fragments (f32 16x4 layout: M = lane%16, K = v + 2*(lane>=16))
            const int am = lane & 15;
            const int ak = (lane >> 4) << 1;
            v2f A[4];
            #pragma unroll
            for (int ks = 0; ks < 4; ++ks) {
                A[ks].x = Uhat[am][4*ks + ak + 0];
                A[ks].y = Uhat[am][4*ks + ak + 1];
            }

            const int shift   = 3 * g;
            const int lowmask = (1 << shift) - 1;

            // 32 column tiles of 16; wave w owns tiles {w, w+8, w+16, w+24}
            for (int t = 0; t < 4; ++t) {
                const int col  = (wave + 8*t) * 16 + (lane & 15);
                // amp index with the 3 group bits zeroed
                const int amp0 = (col & lowmask) | ((col >> shift) << (shift + 3));

                v8f acc = {0.f,0.f,0.f,0.f,0.f,0.f,0.f,0.f};
                #pragma unroll
                for (int ks = 0; ks < 4; ++ks) {
                    // B fragment (4x16, mirrored layout: K = v + 2*(lane>=16))
                    const int kb0 = 4*ks + ((lane >> 4) << 1);
                    const int kb1 = kb0 + 1;
                    v2f B;
                    B.x = S[((kb0 >> 3) << 12) | amp0 | ((kb0 & 7) << shift)];
                    B.y = S[((kb1 >> 3) << 12) | amp0 | ((kb1 & 7) << shift)];
                    acc = __builtin_amdgcn_wmma_f32_16x16x4_f32(
                              false, A[ks], false, B, (short)0, acc, false, false);
                }
                // D writeback (M = v + 8*(lane>=16), N = lane%16); tile owns
                // its 16 columns exclusively -> in-place safe.
                const int mbase = (lane >> 4) << 3;
                #pragma unroll
                for (int v = 0; v < 8; ++v) {
                    const int M = mbase + v;
                    S[((M >> 3) << 12) | amp0 | ((M & 7) << shift)] = acc[v];
                }
            }
            __syncthreads();
        }

        // ---- CX ring as one basis permutation: y_j = XOR_{i>=j} x_i (j<11),
        //      y_11 = XOR_{i<11} x_i.  Read-all / barrier / scatter-all. ----
        {
            float pr[16], pi[16];
            int   ys[16];
            #pragma unroll
            for (int k = 0; k < 16; ++k) {
                const int x = tid + THREADS * k;
                pr[k] = S[x];
                pi[k] = S[NAMP + x];
                int p = x;
                p ^= p >> 1; p ^= p >> 2; p ^= p >> 4; p ^= p >> 8;  // suffix XOR
                ys[k] = (p & 0x7FF) | ((((p & 1) ^ ((x >> 11) & 1))) << 11);
            }
            __syncthreads();
            #pragma unroll
            for (int k = 0; k < 16; ++k) {
                S[ys[k]]        = pr[k];
                S[NAMP + ys[k]] = pi[k];
            }
            __syncthreads();
        }
    }

    // ---- <Z_i> = sum_x |amp(x)|^2 * (-1)^{bit_{11-i}(x)} ----
    if (tid < NQ) zacc[tid] = 0.0f;
    __syncthreads();
    {
        float part[NQ];
        #pragma unroll
        for (int j = 0; j < NQ; ++j) part[j] = 0.0f;
        #pragma unroll
        for (int k = 0; k < 16; ++k) {
            const int x = tid + THREADS * k;
            const float re = S[x], im = S[NAMP + x];
            const float p = re*re + im*im;
            #pragma unroll
            for (int j = 0; j < NQ; ++j)
                part[j] += ((x >> j) & 1) ? -p : p;
        }
        #pragma unroll
        for (int j = 0; j < NQ; ++j)
            atomicAdd(&zacc[j], part[j]);       // ds_add_f32
    }
    __syncthreads();
    if (tid < NQ)
        out[b * NQ + tid] = zacc[11 - tid];     // my bit j <-> ref qubit 11-j
}

extern "C" void kernel_launch(void* const* d_in, const int* in_sizes, int n_in,
                              void* d_out, int out_size, void* d_ws, size_t ws_size,
                              hipStream_t stream) {
    const float* weights = (const float*)d_in[0];  // 324 floats
    const float* patches = (const float*)d_in[1];  // 1024*12*9 floats
    float* out = (float*)d_out;                    // 1024*12 floats
    (void)in_sizes; (void)n_in; (void)out_size; (void)d_ws; (void)ws_size;

    qcircuit_sim_kernel<<<1024, THREADS, 0, stream>>>(weights, patches, out);
}